// GConvLSTM_decoder_6794638262635
// MI455X (gfx1250) — compile-verified
//
#include <hip/hip_runtime.h>
#include <hip/hip_bf16.h>
#include <math.h>
#include <stdint.h>

// ---------------------------------------------------------------------------
// GConvLSTM decoder for MI455X (gfx1250, wave32, WMMA).
// - Dense math: V_WMMA_F32_16X16X4_F32 (exact fp32; compute is free here)
// - LDS staging: GLOBAL_LOAD_ASYNC_TO_LDS_B128 (ASYNCcnt) with double buffer
// - Edge aggregate: wave-per-edge float4 gathers + global_atomic_add_f32 (L2)
// ---------------------------------------------------------------------------

typedef __attribute__((ext_vector_type(2))) float v2f;
typedef __attribute__((ext_vector_type(8))) float v8f;

#define HID 128
#define OUTF 128

// -------- CDNA5 async global->LDS copy (inline asm, ISA 15.18.3) -----------

__device__ __forceinline__ void async_b128_to_lds(unsigned lds_off, const float* gaddr) {
    asm volatile("global_load_async_to_lds_b128 %0, %1, off"
                 :: "v"(lds_off), "v"(gaddr)
                 : "memory");
}

__device__ __forceinline__ void wait_async0() {
    asm volatile("s_wait_asynccnt 0x0" ::: "memory");
}

// ---------------- small elementwise / edge kernels -------------------------

__global__ void k_fill1(float* __restrict__ deg, int n) {
    int i = blockIdx.x * blockDim.x + threadIdx.x;
    if (i < n) deg[i] = 1.0f;
}

__global__ void k_deg_add(float* __restrict__ deg,
                          const int* __restrict__ dst,
                          const float* __restrict__ ea, int e) {
    int i = blockIdx.x * blockDim.x + threadIdx.x;
    if (i < e) unsafeAtomicAdd(&deg[dst[i]], ea[i]);
}

__global__ void k_rsqrt_inplace(float* __restrict__ d, int n) {
    int i = blockIdx.x * blockDim.x + threadIdx.x;
    if (i < n) d[i] = rsqrtf(d[i]);
}

__global__ void k_edge_norm(float* __restrict__ norm,
                            const float* __restrict__ dinv,
                            const int* __restrict__ src,
                            const int* __restrict__ dst,
                            const float* __restrict__ ea, int e) {
    int i = blockIdx.x * blockDim.x + threadIdx.x;
    if (i < e) norm[i] = dinv[src[i]] * ea[i] * dinv[dst[i]];
}

// ---------------- WMMA GEMM + self-loop fusion -----------------------------
// X = [X0 | X1] (two [n,128] halves), W is [256, F] row-major.
// Writes XW (raw product) and AGG = XW * dinv(row)^2  (GCN self-loop term).
// Block: 256 thr = 8 waves; block tile 128(M) x 16(N); K in 8 chunks of 32,
// staged into double-buffered LDS via async-to-LDS so fill overlaps WMMA.

__global__ __launch_bounds__(256) void k_gemm_gcn(
    const float* __restrict__ X0, const float* __restrict__ X1,
    const float* __restrict__ W,  const float* __restrict__ dinv,
    float* __restrict__ XW, float* __restrict__ AGG,
    int nrows, int F)
{
    __shared__ __align__(16) float lds[2][128 * 36];  // stride 36 -> 16B rows, b64-aligned pairs

    const int lane = threadIdx.x & 31;
    const int wave = threadIdx.x >> 5;
    const int half = lane >> 4;      // 0: K pair {0,1}, 1: K pair {2,3}
    const int l16  = lane & 15;
    const int m0   = blockIdx.x * 128;
    const int n0   = blockIdx.y * 16;
    const int col  = n0 + l16;
    const int mrow = wave * 16 + l16;

    v8f acc = {};

    auto stage = [&](int ch, int buf) {
        const int c0  = ch * 32;
        const float* Xh = (c0 < 128) ? X0 : X1;
        const int c0h = (c0 < 128) ? c0 : (c0 - 128);
        #pragma unroll
        for (int i = 0; i < 4; ++i) {                 // 128x8 float4s / 256 thr
            int idx = threadIdx.x + i * 256;
            int r   = idx >> 3;
            int c4  = idx & 7;
            int gr  = m0 + r;
            if (gr < nrows) {
                unsigned loff = (unsigned)(uintptr_t)&lds[buf][r * 36 + c4 * 4];
                async_b128_to_lds(loff, Xh + (size_t)gr * HID + c0h + c4 * 4);
            }
        }
    };

    stage(0, 0);

    #pragma unroll 1
    for (int ch = 0; ch < 8; ++ch) {
        wait_async0();          // my async fills done
        __syncthreads();        // everyone's fills done; prior readers done
        if (ch + 1 < 8) stage(ch + 1, (ch + 1) & 1);

        const float* lb = &lds[ch & 1][0];
        const v2f*   ap = (const v2f*)(lb + mrow * 36 + half * 2);
        const float* wp = W + (size_t)(ch * 32 + half * 2) * F + col;

        #pragma unroll
        for (int kk = 0; kk < 8; ++kk) {              // K step of 4
            v2f a = ap[kk * 2];                       // ds_load (pairs fuse to 2addr_b64)
            v2f b;
            b.x = wp[0];
            b.y = wp[(size_t)F];
            wp += (size_t)4 * F;
            acc = __builtin_amdgcn_wmma_f32_16x16x4_f32(
                      false, a, false, b, (short)0, acc, false, false);
        }
    }

    // C/D layout: VGPR r -> lanes0-15: M=r, lanes16-31: M=r+8
    #pragma unroll
    for (int r = 0; r < 8; ++r) {
        int row = m0 + wave * 16 + r + half * 8;
        if (row < nrows) {
            float di = dinv[row];
            float v  = acc[r];
            size_t idx = (size_t)row * F + col;
            XW[idx]  = v;
            AGG[idx] = v * di * di;
        }
    }
}

// ---------------- edge scatter: AGG[dst] += XW[src] * norm -----------------
// One wave32 per edge; float4 gathers (global_load_b128) + f32 atomics to L2.
// XW/AGG are 41 MB each -> resident in the 192 MB L2, so this streams at L2 rate.

__global__ __launch_bounds__(256) void k_scatter(
    const float* __restrict__ XW, float* __restrict__ AGG,
    const int* __restrict__ src, const int* __restrict__ dst,
    const float* __restrict__ norm, int e, int F)
{
    int gw   = (blockIdx.x * blockDim.x + threadIdx.x) >> 5;
    int lane = threadIdx.x & 31;
    if (gw >= e) return;
    int   s  = src[gw];
    int   d  = dst[gw];
    float nm = norm[gw];
    const float4* xs = (const float4*)(XW + (size_t)s * F);
    float* ad = AGG + (size_t)d * F;
    int nf4 = F >> 2;
    for (int i = lane; i < nf4; i += 32) {
        float4 v = xs[i];
        int b = i * 4;
        unsafeAtomicAdd(ad + b + 0, v.x * nm);
        unsafeAtomicAdd(ad + b + 1, v.y * nm);
        unsafeAtomicAdd(ad + b + 2, v.z * nm);
        unsafeAtomicAdd(ad + b + 3, v.w * nm);
    }
}

// ---------------- init state: elu(AGG + b) -> h0, c ------------------------

__global__ void k_init_state(const float* __restrict__ AGG,
                             const float* __restrict__ b,
                             float* __restrict__ h0, float* __restrict__ cbuf,
                             int n)
{
    int tid = blockIdx.x * blockDim.x + threadIdx.x;
    if (tid >= n * 256) return;
    int r = tid >> 8, j = tid & 255;
    float v = AGG[(size_t)r * 256 + j] + b[j];
    v = (v > 0.f) ? v : expm1f(v);
    if (j < OUTF) h0[(size_t)r * OUTF + j]            = v;
    else          cbuf[(size_t)r * OUTF + (j - OUTF)] = v;
}

// ---------------- fused LSTM pointwise -------------------------------------

__device__ __forceinline__ float sigm(float x) { return 1.f / (1.f + expf(-x)); }

__global__ void k_lstm(const float* __restrict__ AGG,
                       const float* __restrict__ bc,
                       float* __restrict__ cbuf, float* __restrict__ hout,
                       int n)
{
    int tid = blockIdx.x * blockDim.x + threadIdx.x;
    if (tid >= n * OUTF) return;
    int r = tid >> 7, j = tid & 127;
    size_t base = (size_t)r * 512;
    float vi = AGG[base +   0 + j] + bc[  0 + j];
    float vf = AGG[base + 128 + j] + bc[128 + j];
    float vo = AGG[base + 256 + j] + bc[256 + j];
    float vg = AGG[base + 384 + j] + bc[384 + j];
    float cn = sigm(vf) * cbuf[tid] + sigm(vi) * tanhf(vg);
    cbuf[tid] = cn;
    hout[tid] = sigm(vo) * tanhf(cn);
}

// ---------------------------------------------------------------------------

extern "C" void kernel_launch(void* const* d_in, const int* in_sizes, int n_in,
                              void* d_out, int out_size, void* d_ws, size_t ws_size,
                              hipStream_t stream)
{
    const float* h   = (const float*)d_in[0];
    const float* c   = (const float*)d_in[1];
    const int*   ei  = (const int*)d_in[2];
    const float* ea  = (const float*)d_in[3];
    const float* Wi  = (const float*)d_in[4];
    const float* bi  = (const float*)d_in[5];
    const float* Wc  = (const float*)d_in[6];
    const float* bc  = (const float*)d_in[7];

    const int n = in_sizes[0] / HID;     // 20000
    const int e = in_sizes[3];           // 320000
    const int* src = ei;
    const int* dst = ei + e;
    float* out = (float*)d_out;

    // workspace carving (256B aligned)
    char* ws = (char*)d_ws;
    size_t off = 0;
    auto carve = [&](size_t bytes) -> char* {
        size_t p = off;
        off += (bytes + 255) & ~(size_t)255;
        return ws + p;
    };
    float* XW   = (float*)carve((size_t)n * 512 * 4);
    float* AGG  = (float*)carve((size_t)n * 512 * 4);
    float* dinv = (float*)carve((size_t)n * 4);
    float* nrm  = (float*)carve((size_t)e * 4);
    float* h0   = (float*)carve((size_t)n * OUTF * 4);
    float* cbuf = (float*)carve((size_t)n * OUTF * 4);
    (void)ws_size;

    const int B = 256;
    // degree -> dinv -> edge norm
    k_fill1        <<<(n + B - 1) / B, B, 0, stream>>>(dinv, n);
    k_deg_add      <<<(e + B - 1) / B, B, 0, stream>>>(dinv, dst, ea, e);
    k_rsqrt_inplace<<<(n + B - 1) / B, B, 0, stream>>>(dinv, n);
    k_edge_norm    <<<(e + B - 1) / B, B, 0, stream>>>(nrm, dinv, src, dst, ea, e);

    const int gm = (n + 127) / 128;
    const int scat_grid = (int)(((size_t)e * 32 + B - 1) / B);

    // init GCN: concat(h,c) @ W_init  (F=256), scatter, elu split
    {
        dim3 g(gm, 256 / 16);
        k_gemm_gcn<<<g, B, 0, stream>>>(h, c, Wi, dinv, XW, AGG, n, 256);
        k_scatter <<<scat_grid, B, 0, stream>>>(XW, AGG, src, dst, nrm, e, 256);
        k_init_state<<<((size_t)n * 256 + B - 1) / B, B, 0, stream>>>(AGG, bi, h0, cbuf, n);
    }

    // 8 LSTM steps: concat(h, h_cur) @ W_cell (F=512); h_next lands in out[t]
    for (int t = 0; t < 8; ++t) {
        const float* hcur = (t == 0) ? (const float*)h0
                                     : (const float*)(out + (size_t)(t - 1) * n * OUTF);
        dim3 g(gm, 512 / 16);
        k_gemm_gcn<<<g, B, 0, stream>>>(h, hcur, Wc, dinv, XW, AGG, n, 512);
        k_scatter <<<scat_grid, B, 0, stream>>>(XW, AGG, src, dst, nrm, e, 512);
        k_lstm    <<<((size_t)n * OUTF + B - 1) / B, B, 0, stream>>>(
                      AGG, bc, cbuf, out + (size_t)t * n * OUTF, n);
    }
}